// FeaturePnP_3332894622635
// MI455X (gfx1250) — compile-verified
//
#include <hip/hip_runtime.h>
#include <math.h>

#define CCH 1024
#define HH 128
#define WW 128
#define NPTS 4096
#define HWIMG (HH * WW)
#define HWC (HH * WW * CCH)
#define INV_SR (1.0f / 8.0f)

// state layout (floats) inside workspace
#define ST_R 0      // 9
#define ST_T 9      // 3
#define ST_RC 12    // 9  (candidate R)
#define ST_TC 21    // 3  (candidate t)
#define ST_LAM 24
#define ST_PREV 25
#define ST_COST 26
#define ST_H 32     // 36
#define ST_G 68     // 6

#define ROWSTRIDE 16  // padded row stride for staged D / A matrices

typedef __attribute__((ext_vector_type(2))) float v2f;
typedef __attribute__((ext_vector_type(8))) float v8f;

__device__ __forceinline__ int iclamp(int v, int lo, int hi) {
  return v < lo ? lo : (v > hi ? hi : v);
}

__device__ __forceinline__ float wave_sum(float v) {
  for (int o = 16; o > 0; o >>= 1) v += __shfl_xor(v, o, 32);
  return v;
}

// Bilinear setup for HWC images: offsets are element offsets of channel-row starts.
__device__ __forceinline__ void bilin_hwc(float x, float y,
                                          int& o00, int& o10, int& o01, int& o11,
                                          float& w00, float& w10, float& w01, float& w11) {
  float x0f = floorf(x), y0f = floorf(y);
  float wx = x - x0f, wy = y - y0f;
  int x0 = iclamp((int)x0f, 0, WW - 1);
  int x1 = iclamp(x0 + 1, 0, WW - 1);
  int y0 = iclamp((int)y0f, 0, HH - 1);
  int y1 = iclamp(y0 + 1, 0, HH - 1);
  o00 = (y0 * WW + x0) * CCH;
  o10 = (y0 * WW + x1) * CCH;
  o01 = (y1 * WW + x0) * CCH;
  o11 = (y1 * WW + x1) * CCH;
  w00 = (1.f - wx) * (1.f - wy);
  w10 = wx * (1.f - wy);
  w01 = (1.f - wx) * wy;
  w11 = wx * wy;
}

__device__ __forceinline__ void project_pt(const float* R, const float* t, const float* K1,
                                           float q0, float q1, float q2,
                                           float& X, float& Y, float& Z, float& u, float& v) {
  X = R[0] * q0 + R[1] * q1 + R[2] * q2 + t[0];
  Y = R[3] * q0 + R[4] * q1 + R[5] * q2 + t[1];
  Z = R[6] * q0 + R[7] * q1 + R[8] * q2 + t[2];
  float fx = K1[0], cx = K1[2], fy = K1[4], cy = K1[5];
  float iz = 1.f / Z;
  u = fx * X * iz + cx;
  v = fy * Y * iz + cy;
}

// ---------------- init: invert r (4x4), relative = q @ r^-1, set pose state ----------------
__global__ void k_init0(const float* __restrict__ q, const float* __restrict__ r,
                        float* __restrict__ st) {
  float a[4][8];
  for (int i = 0; i < 4; i++)
    for (int j = 0; j < 4; j++) {
      a[i][j] = r[i * 4 + j];
      a[i][j + 4] = (i == j) ? 1.f : 0.f;
    }
  for (int col = 0; col < 4; col++) {
    int p = col;
    float best = fabsf(a[col][col]);
    for (int rr = col + 1; rr < 4; rr++) {
      float vv = fabsf(a[rr][col]);
      if (vv > best) { best = vv; p = rr; }
    }
    if (p != col)
      for (int j = 0; j < 8; j++) { float tmp = a[col][j]; a[col][j] = a[p][j]; a[p][j] = tmp; }
    float inv = 1.f / a[col][col];
    for (int j = 0; j < 8; j++) a[col][j] *= inv;
    for (int rr = 0; rr < 4; rr++)
      if (rr != col) {
        float f = a[rr][col];
        for (int j = 0; j < 8; j++) a[rr][j] -= f * a[col][j];
      }
  }
  float rel[4][4];
  for (int i = 0; i < 4; i++)
    for (int j = 0; j < 4; j++) {
      float s = 0.f;
      for (int k = 0; k < 4; k++) s += q[i * 4 + k] * a[k][j + 4];
      rel[i][j] = s;
    }
  for (int i = 0; i < 3; i++)
    for (int j = 0; j < 3; j++) {
      st[ST_R + i * 3 + j] = rel[i][j];
      st[ST_RC + i * 3 + j] = rel[i][j];
    }
  for (int i = 0; i < 3; i++) { st[ST_T + i] = rel[i][3]; st[ST_TC + i] = rel[i][3]; }
  st[ST_LAM] = 0.01f;
  st[ST_PREV] = 3.0e38f;
  st[ST_COST] = 0.f;
}

// ---------------- pts3d0 = from_homog(to_homog(pts3d) @ r^T) ----------------
__global__ void k_pts3d0(const float* __restrict__ p3, const float* __restrict__ r,
                         float* __restrict__ out) {
  int n = blockIdx.x * blockDim.x + threadIdx.x;
  if (n >= NPTS) return;
  float x = p3[n * 3], y = p3[n * 3 + 1], z = p3[n * 3 + 2];
  float o[4];
  for (int i = 0; i < 4; i++)
    o[i] = r[i * 4] * x + r[i * 4 + 1] * y + r[i * 4 + 2] * z + r[i * 4 + 3];
  float inv = 1.f / o[3];
  out[n * 3] = o[0] * inv;
  out[n * 3 + 1] = o[1] * inv;
  out[n * 3 + 2] = o[2] * inv;
}

// ---------------- Sobel (cross-correlation, SAME/zero-pad) + CHW->HWC transpose ----------------
__global__ void k_sobel(const float* __restrict__ img, float* __restrict__ outc,
                        float* __restrict__ outgx, float* __restrict__ outgy) {
  long long tid = (long long)blockIdx.x * blockDim.x + threadIdx.x;
  if (tid >= (long long)HWC) return;
  int c = (int)(tid % CCH);
  int xy = (int)(tid / CCH);
  int x = xy % WW;
  int y = xy / WW;
  const float* base = img + (size_t)c * HWIMG;
  auto at = [&](int yy, int xx) -> float {
    return (yy < 0 || yy >= HH || xx < 0 || xx >= WW) ? 0.f : base[yy * WW + xx];
  };
  float m00 = at(y - 1, x - 1), m01 = at(y - 1, x), m02 = at(y - 1, x + 1);
  float m10 = at(y, x - 1), m12 = at(y, x + 1);
  float m20 = at(y + 1, x - 1), m21 = at(y + 1, x), m22 = at(y + 1, x + 1);
  outc[tid] = base[y * WW + x];
  outgx[tid] = (-m00 + m02 - 2.f * m10 + 2.f * m12 - m20 + m22) * 0.125f;
  outgy[tid] = (-m00 - 2.f * m01 - m02 + m20 + 2.f * m21 + m22) * 0.125f;
}

// ---------------- feat0 = bilinear(imgf0 CHW, points_2D/8), layout [n][c] ----------------
__global__ void k_feat0(const float* __restrict__ img0, const float* __restrict__ pts2d,
                        float* __restrict__ feat0) {
  int wid = threadIdx.x >> 5, lane = threadIdx.x & 31;
  int n = blockIdx.x * (blockDim.x >> 5) + wid;
  if (n >= NPTS) return;
  float x = pts2d[n * 2] * INV_SR, y = pts2d[n * 2 + 1] * INV_SR;
  float x0f = floorf(x), y0f = floorf(y);
  float wx = x - x0f, wy = y - y0f;
  int x0 = iclamp((int)x0f, 0, WW - 1), x1 = iclamp(x0 + 1, 0, WW - 1);
  int y0 = iclamp((int)y0f, 0, HH - 1), y1 = iclamp(y0 + 1, 0, HH - 1);
  int o00 = y0 * WW + x0, o10 = y0 * WW + x1, o01 = y1 * WW + x0, o11 = y1 * WW + x1;
  float w00 = (1.f - wx) * (1.f - wy), w10 = wx * (1.f - wy);
  float w01 = (1.f - wx) * wy, w11 = wx * wy;
  for (int c = lane; c < CCH; c += 32) {
    const float* b = img0 + (size_t)c * HWIMG;
    feat0[(size_t)n * CCH + c] = w00 * b[o00] + w10 * b[o10] + w01 * b[o01] + w11 * b[o11];
  }
}

// ---------------- stage: per-point channel sums -> D(8192x16) and A(8192x16) rows ----------------
__global__ void k_stage(const float* __restrict__ img1h, const float* __restrict__ gxh,
                        const float* __restrict__ gyh, const float* __restrict__ feat0,
                        const float* __restrict__ pts3d0, const float* __restrict__ K1,
                        const float* __restrict__ st, float* __restrict__ Dmat,
                        float* __restrict__ Amat) {
  int wid = threadIdx.x >> 5, lane = threadIdx.x & 31;
  int n = blockIdx.x * (blockDim.x >> 5) + wid;
  if (n >= NPTS) return;
  float R[9], t[3];
  for (int i = 0; i < 9; i++) R[i] = st[ST_R + i];
  for (int i = 0; i < 3; i++) t[i] = st[ST_T + i];
  float q0 = pts3d0[n * 3], q1 = pts3d0[n * 3 + 1], q2 = pts3d0[n * 3 + 2];
  float X, Y, Z, u, v;
  project_pt(R, t, K1, q0, q1, q2, X, Y, Z, u, v);
  float ix = fminf(fmaxf(u * INV_SR, 0.f), (float)(WW - 1));
  float iy = fminf(fmaxf(v * INV_SR, 0.f), (float)(HH - 1));
  int o00, o10, o01, o11;
  float w00, w10, w01, w11;
  bilin_hwc(ix, iy, o00, o10, o01, o11, w00, w10, w01, w11);
  float sxx = 0.f, sxy = 0.f, syy = 0.f, sxe = 0.f, sye = 0.f;
  const float* f0 = feat0 + (size_t)n * CCH;
  // float4 (b128) channel loop: lane handles 4 consecutive channels per 128-ch chunk
  for (int c = lane * 4; c < CCH; c += 128) {
    float4 fa = *(const float4*)(img1h + o00 + c);
    float4 fb = *(const float4*)(img1h + o10 + c);
    float4 fc = *(const float4*)(img1h + o01 + c);
    float4 fd = *(const float4*)(img1h + o11 + c);
    float4 xa = *(const float4*)(gxh + o00 + c);
    float4 xb = *(const float4*)(gxh + o10 + c);
    float4 xc = *(const float4*)(gxh + o01 + c);
    float4 xd = *(const float4*)(gxh + o11 + c);
    float4 ya = *(const float4*)(gyh + o00 + c);
    float4 yb = *(const float4*)(gyh + o10 + c);
    float4 yc = *(const float4*)(gyh + o01 + c);
    float4 yd = *(const float4*)(gyh + o11 + c);
    float4 f0v = *(const float4*)(f0 + c);
#define ACC(W)                                                                     \
    {                                                                              \
      float f1 = w00 * fa.W + w10 * fb.W + w01 * fc.W + w11 * fd.W;                \
      float gx = w00 * xa.W + w10 * xb.W + w01 * xc.W + w11 * xd.W;                \
      float gy = w00 * ya.W + w10 * yb.W + w01 * yc.W + w11 * yd.W;                \
      float e = f1 - f0v.W;                                                        \
      sxx += gx * gx; sxy += gx * gy; syy += gy * gy;                              \
      sxe += gx * e;  sye += gy * e;                                               \
    }
    ACC(x) ACC(y) ACC(z) ACC(w)
#undef ACC
  }
  sxx = wave_sum(sxx); sxy = wave_sum(sxy); syy = wave_sum(syy);
  sxe = wave_sum(sxe); sye = wave_sum(sye);
  if (lane == 0) {
    float fx = K1[0], fy = K1[4];
    float iz = 1.f / Z;
    float s = iz * INV_SR;
    float a0 = s * fx, a2 = -s * fx * X * iz;
    float b1 = s * fy, b2 = -s * fy * Y * iz;
    // A = J_h_p @ [I | -skew(p3)]
    float A0[6] = {a0, 0.f, a2, a2 * Y, a0 * Z - a2 * X, -a0 * Y};
    float A1[6] = {0.f, b1, b2, b2 * Y - b1 * Z, -b2 * X, b1 * X};
    float* Ar0 = Amat + (size_t)(2 * n) * ROWSTRIDE;
    float* Ar1 = Ar0 + ROWSTRIDE;
    float* Dr0 = Dmat + (size_t)(2 * n) * ROWSTRIDE;
    float* Dr1 = Dr0 + ROWSTRIDE;
    for (int k = 0; k < 6; k++) {
      Ar0[k] = A0[k];
      Ar1[k] = A1[k];
      Dr0[k] = sxx * A0[k] + sxy * A1[k];
      Dr1[k] = sxy * A0[k] + syy * A1[k];
    }
    Dr0[6] = sxe;
    Dr1[6] = sye;
    for (int k = 6; k < ROWSTRIDE; k++) { Ar0[k] = 0.f; Ar1[k] = 0.f; }
    for (int k = 7; k < ROWSTRIDE; k++) { Dr0[k] = 0.f; Dr1[k] = 0.f; }
  }
}

// ---------------- WMMA reduction: [H | g] = D^T(7x8192) x A(8192x6) ----------------
__global__ void __launch_bounds__(256) k_wmma(const float* __restrict__ Dmat,
                                              const float* __restrict__ Amat,
                                              float* __restrict__ st) {
  int wid = threadIdx.x >> 5, lane = threadIdx.x & 31;
  int m = lane & 15;        // A: row of A-operand / B: column index; C/D: column N
  int hi = lane >> 4;       // selects K 0..1 vs 2..3 per ISA f32 operand layout
  v8f acc = {};
  const int rows_per_wave = (2 * NPTS) / 8;  // 1024
  int base = wid * rows_per_wave;
  // Unconditional, fully padded loads: rows are 16 floats; cols >= 7 (D) / >= 6 (A) are zero.
  const float* Dp = Dmat + (size_t)(base + 2 * hi) * ROWSTRIDE + m;
  const float* Ap = Amat + (size_t)(base + 2 * hi) * ROWSTRIDE + m;
  for (int p0 = 0; p0 < rows_per_wave; p0 += 4) {
    v2f a, b;
    a.x = Dp[0];              // A[m][k] = D[p0+k][m]
    a.y = Dp[ROWSTRIDE];
    b.x = Ap[0];              // B[k][n] = A[p0+k][n]
    b.y = Ap[ROWSTRIDE];
    acc = __builtin_amdgcn_wmma_f32_16x16x4_f32(false, a, false, b, (short)0, acc,
                                                false, false);
    Dp += 4 * ROWSTRIDE;
    Ap += 4 * ROWSTRIDE;
  }
  __shared__ float red[2048];
  for (int r = 0; r < 8; r++) red[wid * 256 + lane * 8 + r] = acc[r];
  __syncthreads();
  if (wid == 0) {
    for (int r = 0; r < 8; r++) {
      float s = 0.f;
      for (int w = 0; w < 8; w++) s += red[w * 256 + lane * 8 + r];
      int M = r + 8 * hi;   // output row (column of D = index into [H rows | g])
      int Nc = m;           // output col (column of A)
      if (M < 6 && Nc < 6) st[ST_H + M * 6 + Nc] = s;
      else if (M == 6 && Nc < 6) st[ST_G + Nc] = s;
    }
  }
}

// ---------------- solve: LM damping, 6x6 solve, so3exp, candidate pose ----------------
__global__ void k_solve(float* __restrict__ st) {
  float lam = st[ST_LAM];
  float Hd[6][7];
  for (int i = 0; i < 6; i++) {
    for (int j = 0; j < 6; j++) Hd[i][j] = st[ST_H + i * 6 + j];
    Hd[i][i] += (st[ST_H + i * 6 + i] + 1e-9f) * lam;
    Hd[i][6] = -st[ST_G + i];  // solve Hd x = -g
  }
  for (int col = 0; col < 6; col++) {
    int p = col;
    float best = fabsf(Hd[col][col]);
    for (int rr = col + 1; rr < 6; rr++) {
      float vv = fabsf(Hd[rr][col]);
      if (vv > best) { best = vv; p = rr; }
    }
    if (p != col)
      for (int j = col; j < 7; j++) { float tmp = Hd[col][j]; Hd[col][j] = Hd[p][j]; Hd[p][j] = tmp; }
    float inv = 1.f / Hd[col][col];
    for (int j = col; j < 7; j++) Hd[col][j] *= inv;
    for (int rr = 0; rr < 6; rr++)
      if (rr != col) {
        float f = Hd[rr][col];
        for (int j = col; j < 7; j++) Hd[rr][j] -= f * Hd[col][j];
      }
  }
  float dt0 = Hd[0][6], dt1 = Hd[1][6], dt2 = Hd[2][6];
  float w0 = Hd[3][6], w1 = Hd[4][6], w2 = Hd[5][6];
  // so3exp
  float th2 = w0 * w0 + w1 * w1 + w2 * w2;
  float th = sqrtf(fmaxf(th2, 1e-24f));
  float A = (th2 < 1e-16f) ? 1.f : (sinf(th) / th);
  float B = (th2 < 1e-16f) ? 0.5f : ((1.f - cosf(th)) / fmaxf(th2, 1e-24f));
  float Kx[9] = {0.f, -w2, w1, w2, 0.f, -w0, -w1, w0, 0.f};
  float K2[9];
  for (int i = 0; i < 3; i++)
    for (int j = 0; j < 3; j++) {
      float s = 0.f;
      for (int k = 0; k < 3; k++) s += Kx[i * 3 + k] * Kx[k * 3 + j];
      K2[i * 3 + j] = s;
    }
  float dr[9];
  for (int i = 0; i < 3; i++)
    for (int j = 0; j < 3; j++)
      dr[i * 3 + j] = ((i == j) ? 1.f : 0.f) + A * Kx[i * 3 + j] + B * K2[i * 3 + j];
  float R[9], t[3];
  for (int i = 0; i < 9; i++) R[i] = st[ST_R + i];
  for (int i = 0; i < 3; i++) t[i] = st[ST_T + i];
  float Rn[9], tn[3];
  for (int i = 0; i < 3; i++)
    for (int j = 0; j < 3; j++) {
      float s = 0.f;
      for (int k = 0; k < 3; k++) s += dr[i * 3 + k] * R[k * 3 + j];
      Rn[i * 3 + j] = s;
    }
  float dtv[3] = {dt0, dt1, dt2};
  for (int i = 0; i < 3; i++) {
    float s = 0.f;
    for (int k = 0; k < 3; k++) s += dr[i * 3 + k] * t[k];
    tn[i] = s + dtv[i];
  }
  for (int i = 0; i < 9; i++) st[ST_RC + i] = Rn[i];
  for (int i = 0; i < 3; i++) st[ST_TC + i] = tn[i];
  st[ST_COST] = 0.f;
}

// ---------------- cost at candidate pose; stores unclipped p2 ----------------
__global__ void k_cost(const float* __restrict__ img1h, const float* __restrict__ feat0,
                       const float* __restrict__ pts3d0, const float* __restrict__ K1,
                       float* __restrict__ st, float* __restrict__ newp2) {
  int wid = threadIdx.x >> 5, lane = threadIdx.x & 31;
  int n = blockIdx.x * (blockDim.x >> 5) + wid;
  if (n >= NPTS) return;
  float R[9], t[3];
  for (int i = 0; i < 9; i++) R[i] = st[ST_RC + i];
  for (int i = 0; i < 3; i++) t[i] = st[ST_TC + i];
  float q0 = pts3d0[n * 3], q1 = pts3d0[n * 3 + 1], q2 = pts3d0[n * 3 + 2];
  float X, Y, Z, u, v;
  project_pt(R, t, K1, q0, q1, q2, X, Y, Z, u, v);
  if (lane == 0) { newp2[2 * n] = u; newp2[2 * n + 1] = v; }
  float ix = fminf(fmaxf(u * INV_SR, 0.f), (float)(WW - 1));
  float iy = fminf(fmaxf(v * INV_SR, 0.f), (float)(HH - 1));
  int o00, o10, o01, o11;
  float w00, w10, w01, w11;
  bilin_hwc(ix, iy, o00, o10, o01, o11, w00, w10, w01, w11);
  float se = 0.f;
  const float* f0 = feat0 + (size_t)n * CCH;
  for (int c = lane * 4; c < CCH; c += 128) {
    float4 fa = *(const float4*)(img1h + o00 + c);
    float4 fb = *(const float4*)(img1h + o10 + c);
    float4 fc = *(const float4*)(img1h + o01 + c);
    float4 fd = *(const float4*)(img1h + o11 + c);
    float4 f0v = *(const float4*)(f0 + c);
#define ACC(W)                                                                     \
    {                                                                              \
      float f1 = w00 * fa.W + w10 * fb.W + w01 * fc.W + w11 * fd.W;                \
      float e = f1 - f0v.W;                                                        \
      se += e * e;                                                                 \
    }
    ACC(x) ACC(y) ACC(z) ACC(w)
#undef ACC
  }
  se = wave_sum(se);
  if (lane == 0) atomicAdd(&st[ST_COST], se);
}

// ---------------- per-point best update ----------------
__global__ void k_best(const float* __restrict__ st, const float* __restrict__ newp2,
                       float* __restrict__ out) {
  int n = blockIdx.x * blockDim.x + threadIdx.x;
  if (n >= NPTS) return;
  float nc = st[ST_COST] * (1.f / NPTS);
  if (nc <= st[ST_PREV]) {
    out[2 * n] = newp2[2 * n];
    out[2 * n + 1] = newp2[2 * n + 1];
  }
}

// ---------------- scalar accept + lambda update ----------------
__global__ void k_update(float* __restrict__ st) {
  float nc = st[ST_COST] * (1.f / NPTS);
  bool acc = nc <= st[ST_PREV];
  if (acc) {
    for (int i = 0; i < 9; i++) st[ST_R + i] = st[ST_RC + i];
    for (int i = 0; i < 3; i++) st[ST_T + i] = st[ST_TC + i];
    st[ST_PREV] = nc;
  }
  float lam = st[ST_LAM] * (acc ? 0.1f : 10.f);
  st[ST_LAM] = fminf(fmaxf(lam, 1e-6f), 100.f);
}

// ---------------- init finalize: best = pts2d_init, prev_cost = cost_init ----------------
__global__ void k_initfinal(float* __restrict__ st, const float* __restrict__ newp2,
                            float* __restrict__ out) {
  int n = blockIdx.x * blockDim.x + threadIdx.x;
  if (n >= NPTS) return;
  out[2 * n] = newp2[2 * n];
  out[2 * n + 1] = newp2[2 * n + 1];
  if (n == 0) st[ST_PREV] = st[ST_COST] * (1.f / NPTS);
}

extern "C" void kernel_launch(void* const* d_in, const int* in_sizes, int n_in,
                              void* d_out, int out_size, void* d_ws, size_t ws_size,
                              hipStream_t stream) {
  (void)in_sizes; (void)n_in; (void)out_size; (void)ws_size;
  const float* imgf0 = (const float*)d_in[0];
  const float* imgf1 = (const float*)d_in[1];
  const float* pts2d = (const float*)d_in[2];
  const float* pts3d = (const float*)d_in[3];
  const float* qm = (const float*)d_in[4];
  const float* rm = (const float*)d_in[5];
  const float* K1 = (const float*)d_in[6];
  float* out = (float*)d_out;

  float* w = (float*)d_ws;
  float* img1h = w;                                       // HWC imgf1
  float* gxh = img1h + (size_t)HWC;                       // HWC gx
  float* gyh = gxh + (size_t)HWC;                         // HWC gy
  float* feat0 = gyh + (size_t)HWC;                       // N*C
  float* pts3d0 = feat0 + (size_t)NPTS * CCH;             // 3N
  float* Dmat = pts3d0 + (size_t)3 * NPTS;                // 2N*16
  float* Amat = Dmat + (size_t)2 * NPTS * ROWSTRIDE;      // 2N*16
  float* newp2 = Amat + (size_t)2 * NPTS * ROWSTRIDE;     // 2N
  float* st = newp2 + (size_t)2 * NPTS;                   // state

  k_init0<<<1, 1, 0, stream>>>(qm, rm, st);
  k_pts3d0<<<(NPTS + 255) / 256, 256, 0, stream>>>(pts3d, rm, pts3d0);
  k_sobel<<<(HWC + 255) / 256, 256, 0, stream>>>(imgf1, img1h, gxh, gyh);
  k_feat0<<<NPTS / 8, 256, 0, stream>>>(imgf0, pts2d, feat0);
  // initial cost/best via candidate == init pose
  k_cost<<<NPTS / 8, 256, 0, stream>>>(img1h, feat0, pts3d0, K1, st, newp2);
  k_initfinal<<<(NPTS + 255) / 256, 256, 0, stream>>>(st, newp2, out);

  for (int it = 0; it < 10; ++it) {
    k_stage<<<NPTS / 8, 256, 0, stream>>>(img1h, gxh, gyh, feat0, pts3d0, K1, st, Dmat, Amat);
    k_wmma<<<1, 256, 0, stream>>>(Dmat, Amat, st);
    k_solve<<<1, 1, 0, stream>>>(st);
    k_cost<<<NPTS / 8, 256, 0, stream>>>(img1h, feat0, pts3d0, K1, st, newp2);
    k_best<<<(NPTS + 255) / 256, 256, 0, stream>>>(st, newp2, out);
    k_update<<<1, 1, 0, stream>>>(st);
  }
}